// GAT_15977278341799
// MI455X (gfx1250) — compile-verified
//
#include <hip/hip_runtime.h>
#include <hip/hip_bf16.h>
#include <math.h>

// ---------------- problem constants (match reference) ----------------
constexpr int N_NODES   = 50000;
constexpr int N_EDGES   = 800000;
constexpr int IN_FEATS  = 256;
constexpr int OUT_FEATS = 32;
constexpr int NUM_HEADS = 8;
constexpr int HF        = NUM_HEADS * OUT_FEATS;   // 256
constexpr float NEG_SLOPE = 0.2f;

typedef float v2f __attribute__((ext_vector_type(2)));
typedef float v8f __attribute__((ext_vector_type(8)));

// Order-preserving float <-> uint mapping for atomic max on floats.
__device__ __forceinline__ unsigned f2ord(float f) {
    unsigned u = __float_as_uint(f);
    return (u & 0x80000000u) ? ~u : (u | 0x80000000u);
}
__device__ __forceinline__ float ord2f(unsigned u) {
    return (u & 0x80000000u) ? __uint_as_float(u & 0x7FFFFFFFu)
                             : __uint_as_float(~u);
}
// f2ord(-inf) == 0x007FFFFF
#define ORD_NEG_INF 0x007FFFFFu

__device__ __forceinline__ float lrelu(float v) {
    return v > 0.0f ? v : NEG_SLOPE * v;
}

// ---------------- kernel 1: init out=bias, m=-inf, s=0 ----------------
__global__ __launch_bounds__(256) void gat_init_kernel(
    float* __restrict__ out, const float* __restrict__ bias,
    unsigned* __restrict__ mord, float* __restrict__ ssum)
{
    int i = blockIdx.x * blockDim.x + threadIdx.x;      // 0 .. N*256-1
    out[i] = bias[i & (HF - 1)];
    if (i < N_NODES * NUM_HEADS) {
        mord[i] = ORD_NEG_INF;
        ssum[i] = 0.0f;
    }
}

// ---------------- kernel 2: h = feat @ W via V_WMMA_F32_16X16X4_F32 ----
// Block = 512 threads = 16 waves. Block handles 16 output rows x 256 cols.
// Wave w -> 16x16 tile at (m0 = blockIdx*16, n0 = w*16). K = 256 (64 steps).
constexpr int LDA = 260;  // padded LDS row stride (dwords): lanes land on distinct banks

__global__ __launch_bounds__(512) void gat_gemm_kernel(
    const float* __restrict__ feat, const float* __restrict__ W,
    float* __restrict__ h)
{
    __shared__ float As[16 * LDA];
    const int m0 = blockIdx.x * 16;

    // cooperatively stage the 16x256 A tile
    for (int i = threadIdx.x; i < 16 * IN_FEATS; i += 512) {
        int r = i >> 8, c = i & 255;
        As[r * LDA + c] = feat[(size_t)(m0 + r) * IN_FEATS + c];
    }
    __syncthreads();

    const int wave  = threadIdx.x >> 5;   // 0..15 -> column tile
    const int lane  = threadIdx.x & 31;
    const int n0    = wave * 16;
    const int lhalf = lane >> 4;          // 0: K pair {k,k+1}, 1: {k+2,k+3}
    const int l16   = lane & 15;          // M (for A) / N (for B)

    v8f acc = {0.f, 0.f, 0.f, 0.f, 0.f, 0.f, 0.f, 0.f};

    for (int k = 0; k < IN_FEATS; k += 4) {
        const int ka = k + lhalf * 2;
        // A 16x4 fragment: lane<16 holds K={k,k+1}, lane>=16 holds K={k+2,k+3}
        v2f a;
        a.x = As[l16 * LDA + ka];
        a.y = As[l16 * LDA + ka + 1];
        // B 4x16 fragment: VGPR0 = rows {k,k+2}, VGPR1 = rows {k+1,k+3}
        v2f b;
        b.x = W[(size_t)ka * HF + n0 + l16];
        b.y = W[(size_t)(ka + 1) * HF + n0 + l16];
        acc = __builtin_amdgcn_wmma_f32_16x16x4_f32(
            /*neg_a=*/false, a, /*neg_b=*/false, b,
            /*c_mod=*/(short)0, acc, /*reuse_a=*/false, /*reuse_b=*/false);
    }

    // D layout: VGPR r -> row m0 + r (+8 for upper half-lanes), col n0 + l16
    const int rbase = m0 + lhalf * 8;
#pragma unroll
    for (int r = 0; r < 8; ++r) {
        h[(size_t)(rbase + r) * HF + n0 + l16] = acc[r];
    }
}

// ---------------- kernel 3: el/er per (node, head) ----------------
__global__ __launch_bounds__(256) void gat_logits_kernel(
    const float* __restrict__ h, const float* __restrict__ attn_l,
    const float* __restrict__ attn_r, float* __restrict__ el,
    float* __restrict__ er)
{
    int t = blockIdx.x * blockDim.x + threadIdx.x;
    if (t >= N_NODES * NUM_HEADS) return;
    int n = t >> 3, hh = t & 7;
    const float* hrow = h + (size_t)n * HF + hh * OUT_FEATS;
    const float* al   = attn_l + hh * OUT_FEATS;
    const float* ar   = attn_r + hh * OUT_FEATS;
    float sl = 0.f, sr = 0.f;
#pragma unroll
    for (int f = 0; f < OUT_FEATS; ++f) {
        float v = hrow[f];
        sl = fmaf(v, al[f], sl);
        sr = fmaf(v, ar[f], sr);
    }
    el[t] = sl;
    er[t] = sr;
}

// ---------------- kernel 4: segment max over dst ----------------
__global__ __launch_bounds__(256) void gat_edge_max_kernel(
    const int* __restrict__ src, const int* __restrict__ dst,
    const float* __restrict__ el, const float* __restrict__ er,
    unsigned* __restrict__ mord)
{
    int t = blockIdx.x * blockDim.x + threadIdx.x;     // E*8 threads exactly
    int e = t >> 3, hh = t & 7;
    int sN = src[e], dN = dst[e];
    float v = lrelu(el[sN * NUM_HEADS + hh] + er[dN * NUM_HEADS + hh]);
    atomicMax(&mord[dN * NUM_HEADS + hh], f2ord(v));
}

// ---------------- kernel 5: segment sum of exp(e - m) -------------
__global__ __launch_bounds__(256) void gat_edge_sum_kernel(
    const int* __restrict__ src, const int* __restrict__ dst,
    const float* __restrict__ el, const float* __restrict__ er,
    const unsigned* __restrict__ mord, float* __restrict__ ssum)
{
    int t = blockIdx.x * blockDim.x + threadIdx.x;
    int e = t >> 3, hh = t & 7;
    int sN = src[e], dN = dst[e];
    int di = dN * NUM_HEADS + hh;
    float v  = lrelu(el[sN * NUM_HEADS + hh] + er[di]);
    float ex = expf(v - ord2f(mord[di]));
    atomicAdd(&ssum[di], ex);
}

// ---------------- kernel 6: weighted scatter aggregation ----------
// One wave32 per edge. Lanes 0..7 compute a[h]; broadcast via shfl.
// All 32 lanes stream the 256-float source row and atomically add into dst.
__global__ __launch_bounds__(256) void gat_edge_agg_kernel(
    const int* __restrict__ src, const int* __restrict__ dst,
    const float* __restrict__ h, const float* __restrict__ el,
    const float* __restrict__ er, const unsigned* __restrict__ mord,
    const float* __restrict__ ssum, float* __restrict__ out)
{
    const int lane = threadIdx.x & 31;
    const int e = blockIdx.x * 8 + (threadIdx.x >> 5);  // E divisible by 8
    int sN = src[e], dN = dst[e];

    float aval = 0.0f;
    if (lane < NUM_HEADS) {
        int di = dN * NUM_HEADS + lane;
        float v  = lrelu(el[sN * NUM_HEADS + lane] + er[di]);
        float ex = expf(v - ord2f(mord[di]));
        aval = ex / ssum[di];
    }

    const float* hrow = h + (size_t)sN * HF;
    float* orow = out + (size_t)dN * HF;
    __builtin_prefetch(hrow, 0, 1);   // global_prefetch_b8

#pragma unroll
    for (int hh = 0; hh < NUM_HEADS; ++hh) {
        float a = __shfl(aval, hh, 32);
        int c = hh * OUT_FEATS + lane;        // coalesced 128B per head
        atomicAdd(&orow[c], hrow[c] * a);     // global_atomic_add_f32
    }
}

// ---------------- host-side launch ----------------
extern "C" void kernel_launch(void* const* d_in, const int* in_sizes, int n_in,
                              void* d_out, int out_size, void* d_ws, size_t ws_size,
                              hipStream_t stream)
{
    const float* feat   = (const float*)d_in[0];
    const float* W      = (const float*)d_in[1];
    const float* attn_l = (const float*)d_in[2];
    const float* attn_r = (const float*)d_in[3];
    const float* bias   = (const float*)d_in[4];
    const int*   src    = (const int*)d_in[5];
    const int*   dst    = (const int*)d_in[6];
    float* out = (float*)d_out;

    // workspace layout (all fp32 / u32), ~58 MB total
    float*    h    = (float*)d_ws;                             // N*256
    float*    el   = h  + (size_t)N_NODES * HF;                // N*8
    float*    er   = el + (size_t)N_NODES * NUM_HEADS;         // N*8
    unsigned* mord = (unsigned*)(er + (size_t)N_NODES * NUM_HEADS); // N*8
    float*    ssum = (float*)(mord + (size_t)N_NODES * NUM_HEADS);  // N*8

    // 1) init output (= bias), max (= -inf), sum (= 0)
    gat_init_kernel<<<(N_NODES * HF) / 256, 256, 0, stream>>>(out, bias, mord, ssum);

    // 2) projection GEMM on the WMMA pipe (N_NODES = 3125 * 16 exactly)
    gat_gemm_kernel<<<N_NODES / 16, 512, 0, stream>>>(feat, W, h);

    // 3) per-node attention logits
    gat_logits_kernel<<<(N_NODES * NUM_HEADS + 255) / 256, 256, 0, stream>>>(
        h, attn_l, attn_r, el, er);

    // 4) edge softmax: segment max
    gat_edge_max_kernel<<<(N_EDGES * NUM_HEADS) / 256, 256, 0, stream>>>(
        src, dst, el, er, mord);

    // 5) edge softmax: segment sum
    gat_edge_sum_kernel<<<(N_EDGES * NUM_HEADS) / 256, 256, 0, stream>>>(
        src, dst, el, er, mord, ssum);

    // 6) weighted scatter-sum aggregation (wave per edge)
    gat_edge_agg_kernel<<<N_EDGES / 8, 256, 0, stream>>>(
        src, dst, h, el, er, mord, ssum, out);
}